// AttentionPool_56573309223705
// MI455X (gfx1250) — compile-verified
//
#include <hip/hip_runtime.h>
#include <math.h>

#define IN_CH 256

typedef __attribute__((ext_vector_type(2))) float v2f;
typedef __attribute__((ext_vector_type(8))) float v8f;

// ---------- helpers ----------
__device__ __forceinline__ unsigned f32_key(float x) {
    unsigned b = __float_as_uint(x);
    return (b & 0x80000000u) ? ~b : (b | 0x80000000u);
}
__device__ __forceinline__ float key_f32(unsigned k) {
    unsigned b = (k & 0x80000000u) ? (k & 0x7FFFFFFFu) : ~k;
    return __uint_as_float(b);
}
__device__ __forceinline__ float lrelu(float x) { return x >= 0.f ? x : 0.2f * x; }

// ---------- kernel 0: zero scratch accumulators ----------
__global__ void k_zero(float* p, int n) {
    int i = blockIdx.x * blockDim.x + threadIdx.x;
    int stride = gridDim.x * blockDim.x;
    for (; i < n; i += stride) p[i] = 0.f;
}

// ---------- kernel 1: q/k GEMV via V_WMMA_F32_16X16X4_F32 ----------
// Each wave owns a 16-node tile. A = x[tile, kk:kk+4] (16x4), B = 4x16 with
// col0 = Wq slice, col1 = Wk slice, cols 2..15 = 0. 64 accumulating WMMAs
// cover K=256. D col0 -> q, D col1 -> k.
__global__ void k_qk_wmma(const float* __restrict__ x,
                          const float* __restrict__ Wq, const float* __restrict__ Wk,
                          const float* __restrict__ bq, const float* __restrict__ bk,
                          float* __restrict__ q, float* __restrict__ k, int n_nodes) {
    __shared__ float wsh[3][IN_CH];   // row 2 = zeros (branchless B build)
    int tid = threadIdx.x;
    wsh[0][tid] = Wq[tid];
    wsh[1][tid] = Wk[tid];
    wsh[2][tid] = 0.f;
    __syncthreads();

    int wave = tid >> 5;
    int lane = tid & 31;
    int tileBase = (blockIdx.x * 8 + wave) * 16;

    int mrow  = lane & 15;                  // M index (A) / N index (B,D)
    int khalf = (lane >> 4) << 1;           // lanes 16-31 hold K+2,K+3
    int node  = tileBase + mrow;
    int nodeC = node < n_nodes ? node : (n_nodes - 1);
    const float* xrow = x + (size_t)nodeC * IN_CH + khalf;

    int col    = lane & 15;
    int colSel = col < 2 ? col : 2;         // 2 -> zero row

    v8f c = {};
    #pragma unroll 4
    for (int kk = 0; kk < IN_CH; kk += 4) {
        v2f a = *(const v2f*)(xrow + kk);                    // A: K=khalf, khalf+1
        v2f b = *(const v2f*)(&wsh[colSel][kk + khalf]);     // B: same K pair
        c = __builtin_amdgcn_wmma_f32_16x16x4_f32(
                /*neg_a=*/false, a, /*neg_b=*/false, b,
                /*c_mod=*/(short)0, c, /*reuse_a=*/false, /*reuse_b=*/false);
    }

    // D layout: VGPR r holds M=r (lanes 0-15) and M=r+8 (lanes 16-31), N=lane&15
    int mbase = tileBase + ((lane >= 16) ? 8 : 0);
    float biasq = *bq, biask = *bk;
    if (col == 0) {
        #pragma unroll
        for (int r = 0; r < 8; ++r) {
            int n = mbase + r;
            if (n < n_nodes) q[n] = c[r] + biasq;
        }
    } else if (col == 1) {
        #pragma unroll
        for (int r = 0; r < 8; ++r) {
            int n = mbase + r;
            if (n < n_nodes) k[n] = c[r] + biask;
        }
    }
}

// ---------- kernel 2: global max of leaky_relu(q[row]*k[col]) ----------
__global__ void k_edge_max(const int* __restrict__ ei,
                           const float* __restrict__ q, const float* __restrict__ k,
                           unsigned* __restrict__ maxkey, int n_edges) {
    float m = -3.402823466e+38f;
    int stride = gridDim.x * blockDim.x;
    for (int e = blockIdx.x * blockDim.x + threadIdx.x; e < n_edges; e += stride) {
        int r = ei[e];
        int c = ei[n_edges + e];
        m = fmaxf(m, lrelu(q[r] * k[c]));
    }
    #pragma unroll
    for (int o = 16; o; o >>= 1) m = fmaxf(m, __shfl_xor(m, o, 32));
    __shared__ float red[8];
    int lane = threadIdx.x & 31, wid = threadIdx.x >> 5;
    if (lane == 0) red[wid] = m;
    __syncthreads();
    if (wid == 0) {
        float v = (lane < 8) ? red[lane] : -3.402823466e+38f;
        #pragma unroll
        for (int o = 4; o; o >>= 1) v = fmaxf(v, __shfl_xor(v, o, 32));
        if (lane == 0) atomicMax(maxkey, f32_key(v));
    }
}

// ---------- kernel 3: u[row] += exp(att - max) ----------
__global__ void k_edge_expsum(const int* __restrict__ ei,
                              const float* __restrict__ q, const float* __restrict__ k,
                              const unsigned* __restrict__ maxkey,
                              float* __restrict__ u, int n_edges) {
    float mx = key_f32(*maxkey);
    int stride = gridDim.x * blockDim.x;
    for (int e = blockIdx.x * blockDim.x + threadIdx.x; e < n_edges; e += stride) {
        int r = ei[e];
        int c = ei[n_edges + e];
        float att = lrelu(q[r] * k[c]);
        atomicAdd(&u[r], expf(att - mx));
    }
}

// ---------- kernel 4: Z = sum(u) ----------
__global__ void k_reduce_sum(const float* __restrict__ u, float* __restrict__ Z, int n) {
    float s = 0.f;
    int stride = gridDim.x * blockDim.x;
    for (int i = blockIdx.x * blockDim.x + threadIdx.x; i < n; i += stride) s += u[i];
    #pragma unroll
    for (int o = 16; o; o >>= 1) s += __shfl_xor(s, o, 32);
    __shared__ float red[8];
    int lane = threadIdx.x & 31, wid = threadIdx.x >> 5;
    if (lane == 0) red[wid] = s;
    __syncthreads();
    if (wid == 0) {
        float v = (lane < 8) ? red[lane] : 0.f;
        #pragma unroll
        for (int o = 4; o; o >>= 1) v += __shfl_xor(v, o, 32);
        if (lane == 0) atomicAdd(Z, v);
    }
}

// ---------- kernel 5: aw = u / Z ----------
__global__ void k_write_aw(const float* __restrict__ u, const float* __restrict__ Z,
                           float* __restrict__ aw, int n) {
    float inv = 1.f / *Z;
    int i = blockIdx.x * blockDim.x + threadIdx.x;
    if (i < n) aw[i] = u[i] * inv;
}

// ---------- kernel 6: y[c] = sum_n aw[n] * x[n,c] ----------
__global__ void k_weighted_x(const float* __restrict__ x, const float* __restrict__ aw,
                             float* __restrict__ y, int n_nodes) {
    const int ROWS = 128;
    int c = threadIdx.x;               // 256 threads = 256 channels, coalesced rows
    int n0 = blockIdx.x * ROWS;
    int n1 = n0 + ROWS; if (n1 > n_nodes) n1 = n_nodes;
    float acc = 0.f;
    for (int n = n0; n < n1; ++n)
        acc = fmaf(aw[n], x[(size_t)n * IN_CH + c], acc);
    atomicAdd(&y[c], acc);
}

// ---------- kernel 7: graph_emb = y @ Wv + bv  (sum(aw)==1 => bias scale 1) ----------
__global__ void k_final_gemv(const float* __restrict__ y, const float* __restrict__ Wv,
                             const float* __restrict__ bv, float* __restrict__ out) {
    __shared__ float ysh[IN_CH];
    int t = threadIdx.x;
    ysh[t] = y[t];
    __syncthreads();
    float acc = bv[t];
    for (int c = 0; c < IN_CH; ++c)
        acc = fmaf(ysh[c], Wv[c * IN_CH + t], acc);
    out[t] = acc;
}

// ---------- host ----------
extern "C" void kernel_launch(void* const* d_in, const int* in_sizes, int n_in,
                              void* d_out, int out_size, void* d_ws, size_t ws_size,
                              hipStream_t stream) {
    const float* x  = (const float*)d_in[0];
    const int*   ei = (const int*)d_in[1];
    const float* Wq = (const float*)d_in[2];
    const float* bq = (const float*)d_in[3];
    const float* Wk = (const float*)d_in[4];
    const float* bk = (const float*)d_in[5];
    const float* Wv = (const float*)d_in[6];
    const float* bv = (const float*)d_in[7];

    int n_nodes = in_sizes[0] / IN_CH;
    int n_edges = in_sizes[1] / 2;

    float*    ws     = (float*)d_ws;
    float*    q      = ws;                         // [n]
    float*    k      = ws + n_nodes;               // [n]
    float*    u      = ws + 2 * (size_t)n_nodes;   // [n]   (zeroed)
    float*    Z      = ws + 3 * (size_t)n_nodes;   // [1]   (zeroed)
    unsigned* maxkey = (unsigned*)(ws + 3 * (size_t)n_nodes + 1); // [1] (zeroed)
    float*    y      = ws + 3 * (size_t)n_nodes + 8;              // [256] (zeroed)

    float* out_emb = (float*)d_out;        // [256]
    float* out_aw  = out_emb + IN_CH;      // [n_nodes]

    // 0) zero u, Z, maxkey, y  (key 0 == f32_key(-inf)-safe floor)
    int nz = n_nodes + 8 + IN_CH;
    k_zero<<<(nz + 255) / 256, 256, 0, stream>>>(u, nz);

    // 1) q/k via WMMA (one pass over x)
    k_qk_wmma<<<(n_nodes + 127) / 128, 256, 0, stream>>>(x, Wq, Wk, bq, bk, q, k, n_nodes);

    // 2) global max over edges
    k_edge_max<<<1024, 256, 0, stream>>>(ei, q, k, maxkey, n_edges);

    // 3) scatter exp(att - max) into u[row]
    k_edge_expsum<<<1024, 256, 0, stream>>>(ei, q, k, maxkey, u, n_edges);

    // 4) Z = sum(u)
    k_reduce_sum<<<128, 256, 0, stream>>>(u, Z, n_nodes);

    // 5) aw = u / Z  -> d_out[256:]
    k_write_aw<<<(n_nodes + 255) / 256, 256, 0, stream>>>(u, Z, out_aw, n_nodes);

    // 6) y = aw^T x  (second pass over x)
    k_weighted_x<<<(n_nodes + 127) / 128, 256, 0, stream>>>(x, out_aw, y, n_nodes);

    // 7) graph_emb = y @ Wv + bv -> d_out[0:256]
    k_final_gemv<<<1, 256, 0, stream>>>(y, Wv, bv, out_emb);
}